// RoIPooling_18399639896534
// MI455X (gfx1250) — compile-verified
//
#include <hip/hip_runtime.h>

#define POOL 7
#define FH 50
#define FW 50
#define FC 1024
#define NT 256          // FC/4 channels-per-thread granularity
#define DEPTH 8         // LDS ring slots (async pipeline depth)

typedef float f4_t __attribute__((ext_vector_type(4)));
typedef int   v4i  __attribute__((ext_vector_type(4)));

typedef __attribute__((address_space(1))) v4i* g4p_t;   // global v4i*
typedef __attribute__((address_space(3))) v4i* l4p_t;   // LDS v4i*

__device__ __forceinline__ void async_ld_b128(const float* gp, unsigned lds_off) {
#if defined(__gfx1250__) && __has_builtin(__builtin_amdgcn_global_load_async_to_lds_b128)
    __builtin_amdgcn_global_load_async_to_lds_b128(
        (g4p_t)(unsigned long long)gp, (l4p_t)(unsigned long long)lds_off, 0, 0);
#else
    asm volatile("global_load_async_to_lds_b128 %0, %1, off"
                 :: "v"(lds_off), "v"(gp) : "memory");
#endif
}

__device__ __forceinline__ void wait_async_6() {
    asm volatile("s_wait_asynccnt 6" ::: "memory");
}
__device__ __forceinline__ void wait_async_0() {
    asm volatile("s_wait_asynccnt 0" ::: "memory");
}
__device__ __forceinline__ void wait_ds_0() {
    asm volatile("s_wait_dscnt 0" ::: "memory");
}

__global__ __launch_bounds__(NT) void roi_maxpool7(
    const float* __restrict__ fm, const float* __restrict__ rois,
    float* __restrict__ out)
{
    __shared__ f4_t sbuf[DEPTH][NT];

    const int tid = threadIdx.x;
    const int bin = blockIdx.x;           // 0..48
    const int roi = blockIdx.y;           // 0..N-1
    const int ph  = bin / POOL;
    const int pw  = bin - ph * POOL;

    const f4_t r = reinterpret_cast<const f4_t*>(rois)[roi];
    const int h0 = (int)(FH * r.x);       // trunc == floor for nonneg
    const int w0 = (int)(FW * r.y);
    const int h1 = (int)(FH * r.z);
    const int w1 = (int)(FW * r.w);
    const int rh = h1 - h0, rw = w1 - w0;
    const int hstep = rh / POOL, wstep = rw / POOL;

    // reference bin-bound logic (grow end if empty & room, else shrink start)
    int s0 = ph * hstep;
    int e0 = (ph < POOL - 1) ? (ph + 1) * hstep : rh;
    bool em = (s0 == e0);
    bool gr = em && (e0 < rh);
    const int sh = (em && !gr) ? s0 - 1 : s0;
    const int eh = gr ? e0 + 1 : e0;

    s0 = pw * wstep;
    e0 = (pw < POOL - 1) ? (pw + 1) * wstep : rw;
    em = (s0 == e0);
    gr = em && (e0 < rw);
    const int sw = (em && !gr) ? s0 - 1 : s0;
    const int ew = gr ? e0 + 1 : e0;

    const int hs = max(0, h0 + sh), he = min(FH, h0 + eh);
    const int ws = max(0, w0 + sw), we = min(FW, w0 + ew);
    const int npix = (he > hs && we > ws) ? (he - hs) * (we - ws) : 0;

    // LDS byte offset of this thread's 16B slot in ring slot 0
    const unsigned lds0 = (unsigned)(unsigned long long)(&sbuf[0][0])
                        + (unsigned)tid * 16u;
    const float* gbase = fm + (unsigned)tid * 4u;

    f4_t acc;
    acc.x = acc.y = acc.z = acc.w = -__builtin_inff();

    int h = hs, w = ws;
    int issued = 0;
    const int pre = (npix < (DEPTH - 1)) ? npix : (DEPTH - 1);
    for (; issued < pre; ++issued) {
        async_ld_b128(gbase + (size_t)(h * FW + w) * FC,
                      lds0 + (unsigned)(issued & (DEPTH - 1)) * (NT * 16u));
        if (++w == we) { w = ws; ++h; }
    }

    for (int p = 0; p < npix; ++p) {
        // loads complete in order: steady state has issued-p-1 == 6 older ops
        if (issued < npix) wait_async_6(); else wait_async_0();

        f4_t v = sbuf[p & (DEPTH - 1)][tid];
        acc.x = fmaxf(acc.x, v.x);
        acc.y = fmaxf(acc.y, v.y);
        acc.z = fmaxf(acc.z, v.z);
        acc.w = fmaxf(acc.w, v.w);

        if (issued < npix) {
            wait_ds_0();  // slot read retired before async engine may overwrite it
            async_ld_b128(gbase + (size_t)(h * FW + w) * FC,
                          lds0 + (unsigned)(issued & (DEPTH - 1)) * (NT * 16u));
            ++issued;
            if (++w == we) { w = ws; ++h; }
        }
    }

    // streamed output, never re-read: keep it out of L2 (feature map lives there)
    f4_t* o = reinterpret_cast<f4_t*>(out) + ((size_t)roi * (POOL * POOL) + bin) * NT + tid;
    __builtin_nontemporal_store(acc, o);
}

extern "C" void kernel_launch(void* const* d_in, const int* in_sizes, int n_in,
                              void* d_out, int out_size, void* d_ws, size_t ws_size,
                              hipStream_t stream) {
    const float* features = (const float*)d_in[0];   // [1,50,50,1024] f32
    const float* rois     = (const float*)d_in[1];   // [N,4] f32
    float* out            = (float*)d_out;           // [N,7,7,1024] f32
    const int N = in_sizes[1] / 4;
    dim3 grid(POOL * POOL, N, 1);
    roi_maxpool7<<<grid, NT, 0, stream>>>(features, rois, out);
}